// RuleAttention_12979391169079
// MI455X (gfx1250) — compile-verified
//
#include <hip/hip_runtime.h>
#include <hip/hip_bf16.h>
#include <stdint.h>

typedef __attribute__((ext_vector_type(2))) float v2f;
typedef __attribute__((ext_vector_type(8))) float v8f;

#define B_SZ 512
#define O_SZ 32
#define R_SZ 2048
#define D_SZ 256

#define QSTR 260      // padded LDS stride for Q rows (floats): 260%64==4 -> conflict-free b64
#define SSTR 2052     // padded LDS stride for score rows (floats)

__device__ __forceinline__ float wred_sum(float v) {
#pragma unroll
  for (int off = 16; off > 0; off >>= 1) v += __shfl_xor(v, off, 32);
  return v;
}

__global__ __launch_bounds__(512, 1)
void RuleAttention_12979391169079_kernel(const float* __restrict__ query,
                                         const float* __restrict__ context,
                                         float* __restrict__ out) {
  extern __shared__ float smem[];
  float* q_lds = smem;                 // 32 * QSTR floats
  float* s_lds = smem + O_SZ * QSTR;   // 32 * SSTR floats

  const int b    = blockIdx.x;
  const int tid  = threadIdx.x;
  const int wave = tid >> 5;           // 0..15
  const int lane = tid & 31;
  const int row16 = lane & 15;
  const int kh    = (lane >> 4) << 1;  // 0 for lanes 0-15, 2 for lanes 16-31

  // ---- Stage Q[b] (32x256) into LDS, padded stride, coalesced ----
  const float* qsrc = query + (size_t)b * O_SZ * D_SZ;
#pragma unroll
  for (int j = 0; j < (O_SZ * D_SZ) / 512; ++j) {
    int idx = tid + 512 * j;
    int r = idx >> 8;        // /256
    int c = idx & 255;
    q_lds[r * QSTR + c] = qsrc[idx];
  }
  __syncthreads();

  // ---- GEMM: scores[32,2048] = Q * C^T via v_wmma_f32_16x16x4_f32 ----
  const float* ctx = context + (size_t)b * R_SZ * D_SZ;
  const float* a0p = &q_lds[row16 * QSTR + kh];
  const float* a1p = &q_lds[(16 + row16) * QSTR + kh];

#pragma unroll 1
  for (int t = 0; t < 8; ++t) {
    const int n0 = (wave + (t << 4)) << 4;               // n-tile base, 0..2032
    const float* bp = ctx + (size_t)(n0 + row16) * D_SZ + kh;

    v8f acc0 = {};  // m-tile 0 (rows 0..15)
    v8f acc1 = {};  // m-tile 1 (rows 16..31)

#pragma unroll 4
    for (int k0 = 0; k0 < D_SZ; k0 += 4) {
      v2f bf = *(const v2f*)(bp + k0);      // B frag: 4x16 f32, K=k0..k0+3
      v2f a0 = *(const v2f*)(a0p + k0);     // A frag m0=0
      v2f a1 = *(const v2f*)(a1p + k0);     // A frag m0=16
      acc0 = __builtin_amdgcn_wmma_f32_16x16x4_f32(
          false, a0, false, bf, (short)0, acc0, false, false);
      acc1 = __builtin_amdgcn_wmma_f32_16x16x4_f32(
          false, a1, false, bf, (short)0, acc1, false, false);
    }

    // C/D layout: VGPR v -> M = v + 8*(lane>=16), N = n0 + (lane&15)
    const int mhi = (lane >> 4) << 3;   // 0 or 8
#pragma unroll
    for (int v = 0; v < 8; ++v) {
      s_lds[(mhi + v) * SSTR + n0 + row16]      = acc0[v];
      s_lds[(16 + mhi + v) * SSTR + n0 + row16] = acc1[v];
    }
  }
  __syncthreads();

  // ---- Sparsemax per row (Michelot): each wave owns 2 rows ----
#pragma unroll 1
  for (int rr = 0; rr < 2; ++rr) {
    const int m = wave * 2 + rr;       // 0..31
    float vals[64];
#pragma unroll
    for (int j = 0; j < 64; ++j) vals[j] = s_lds[m * SSTR + j * 32 + lane];

    // tau_0 = (sum(z) - 1)/K
    float s = 0.f;
#pragma unroll
    for (int j = 0; j < 64; ++j) s += vals[j];
    s = wred_sum(s);
    float tau = (s - 1.0f) * (1.0f / (float)R_SZ);

    // iterate: S = {z > tau}; tau = (sum_S - 1)/|S|  (exact once support stabilizes)
#pragma unroll 1
    for (int it = 0; it < 32; ++it) {
      float sum = 0.f, cnt = 0.f;
#pragma unroll
      for (int j = 0; j < 64; ++j) {
        bool p = vals[j] > tau;
        sum += p ? vals[j] : 0.f;
        cnt += p ? 1.0f : 0.f;
      }
      sum = wred_sum(sum);
      cnt = wred_sum(cnt);
      tau = (sum - 1.0f) / cnt;
    }

    float* orow = out + ((size_t)b * O_SZ + m) * R_SZ;
#pragma unroll
    for (int j = 0; j < 64; ++j)
      orow[j * 32 + lane] = fmaxf(vals[j] - tau, 0.0f);
  }
}

extern "C" void kernel_launch(void* const* d_in, const int* in_sizes, int n_in,
                              void* d_out, int out_size, void* d_ws, size_t ws_size,
                              hipStream_t stream) {
  const float* query   = (const float*)d_in[0];   // [512, 32, 256]
  const float* context = (const float*)d_in[1];   // [512, 2048, 256]
  float* out = (float*)d_out;                     // [512*32, 2048]

  const size_t shmem = (size_t)(O_SZ * QSTR + O_SZ * SSTR) * sizeof(float); // ~289 KB
  RuleAttention_12979391169079_kernel<<<B_SZ, 512, shmem, stream>>>(query, context, out);
}